// OHLCVPackedScaler_28896539967793
// MI455X (gfx1250) — compile-verified
//
#include <hip/hip_runtime.h>
#include <stdint.h>

// ---------------------------------------------------------------------------
// OHLCVPackedScaler for MI455X (gfx1250, wave32)
//
// B=4, S=4096, D=32. Group key = (sample_id in [0,8)) x (group_id in [0,4))
// => 32 bins per batch row. Two streaming passes, no atomics (deterministic):
//   pass 1: per 256-token block, per-bin (cnt, sum, sumsq) partials -> ws
//   pass 2: async-stage the batch's 16 partial tables into LDS
//           (global_load_async_to_lds_b128 + s_wait_asynccnt), reduce in
//           fixed order, finalize loc/scale per token.
// Roofline: ~4.2 MB of input at 23.3 TB/s => ~0.2 us; pure streaming, so the
// win comes from b128 clause loads + async LDS broadcast, not matrix ops.
// ---------------------------------------------------------------------------

#define BATCHES          4
#define SEQ              4096
#define DIM              32
#define NBIN             32                       // 8 sample ids * 4 groups
#define TPB              256                      // 8 wave32 per block
#define BLOCKS_PER_BATCH (SEQ / TPB)              // 16
#define TOTAL_BLOCKS     (BATCHES * BLOCKS_PER_BATCH) // 64
#define BIN_STRIDE       4                        // cnt,sum,sumsq,pad -> 16B
#define TBL_FLOATS       (NBIN * BIN_STRIDE)      // 128 floats / block table

#define AS1 __attribute__((address_space(1)))
#define AS3 __attribute__((address_space(3)))

typedef int v4i __attribute__((vector_size(16)));  // builtin pointee type

// --- CDNA5 async global->LDS copy (16B per lane), ASYNCcnt-tracked ----------
__device__ __forceinline__ void async_copy16(const void* g, void* l) {
#if __has_builtin(__builtin_amdgcn_global_load_async_to_lds_b128)
  // signature: (v4i addrspace(1)*, v4i addrspace(3)*, imm offset, imm cpol)
  __builtin_amdgcn_global_load_async_to_lds_b128(
      (AS1 v4i*)g, (AS3 v4i*)l, 0, 0);
#else
  AS1 v4i* gp = (AS1 v4i*)g;
  AS3 v4i* lp = (AS3 v4i*)l;
  asm volatile("global_load_async_to_lds_b128 %0, %1, off"
               :: "v"(lp), "v"(gp) : "memory");
#endif
}

__device__ __forceinline__ void wait_async0() {
#if __has_builtin(__builtin_amdgcn_s_wait_asynccnt)
  __builtin_amdgcn_s_wait_asynccnt(0);
#else
  asm volatile("s_wait_asynccnt 0" ::: "memory");
#endif
}

__device__ __forceinline__ int group_of(int vid) {
  // variate 0..3 -> 0 (OHLC), 4 -> 1 (vol), 5 -> 2 (min), 6 -> 3 (dow)
  return (vid >= 4) ? (vid - 3) : 0;
}

// ---------------------------------------------------------------------------
// Pass 2 (defined first so its asm leads the disasm snippet):
// async-stage the batch's partial tables to LDS, reduce, finalize.
// ---------------------------------------------------------------------------
__global__ __launch_bounds__(TPB)
void ohlcv_finalize_kernel(const float* __restrict__ ws,
                           const int*   __restrict__ sample_id,
                           const int*   __restrict__ variate_id,
                           float*       __restrict__ out) {
  __shared__ __align__(16) float tbl[BLOCKS_PER_BATCH * TBL_FLOATS]; // 8 KB
  __shared__ float tot[NBIN * 3];

  const int tid = threadIdx.x;
  const int b   = blockIdx.x / BLOCKS_PER_BATCH;
  const float* src = ws + (long)b * BLOCKS_PER_BATCH * TBL_FLOATS;   // 2048 f

  // 512 x 16B async transfers: 2 per thread; each wave waits its ASYNCcnt,
  // the block barrier then publishes LDS to all waves.
#pragma unroll
  for (int r = 0; r < 2; ++r) {
    const int idx = r * TPB + tid;                 // 16B granule 0..511
    async_copy16(src + idx * 4, &tbl[idx * 4]);
  }
  wait_async0();
  __syncthreads();

  // Reduce the 16 block-partials per (bin, field) in fixed blk order.
  if (tid < NBIN * 3) {
    const int g = tid / 3;
    const int k = tid - 3 * g;
    float acc = 0.f;
#pragma unroll
    for (int blk = 0; blk < BLOCKS_PER_BATCH; ++blk) {
      acc += tbl[blk * TBL_FLOATS + g * BIN_STRIDE + k];
    }
    tot[g * 3 + k] = acc;
  }
  __syncthreads();

  // Finalize one token per thread.
  const long t   = (long)blockIdx.x * TPB + tid;
  const int  sid = sample_id[t];
  const int  vid = variate_id[t];
  const int  bin = (sid * 4 + group_of(vid)) & (NBIN - 1);

  const float cnt = tot[bin * 3 + 0];
  const float s1  = tot[bin * 3 + 1];
  const float s2  = tot[bin * 3 + 2];

  // safe_div semantics: denom==0 -> divide by 1
  float loc    = s1 / ((cnt == 0.f) ? 1.f : cnt);
  // sum((t-loc)^2 * obs) == s2 - 2*loc*s1 + loc^2*cnt  (loc const per bin)
  float varnum = fmaf(loc * loc, cnt, fmaf(-2.f * loc, s1, s2));
  varnum       = fmaxf(varnum, 0.f);               // guard fp cancellation
  const float den = cnt - 1.f;                     // CORRECTION = 1
  const float var = varnum / ((den == 0.f) ? 1.f : den);
  float scale  = sqrtf(var + 1e-5f);               // MIN_SCALE

  if (sid == 0) { loc = 0.f; scale = 1.f; }        // padding override

  out[t]                          = loc;           // loc  [B,S,1] flat
  out[(long)BATCHES * SEQ + t]    = scale;         // scale [B,S,1] flat
}

// ---------------------------------------------------------------------------
// Pass 1: per-block per-bin partial moments (deterministic fixed-order scan)
// ---------------------------------------------------------------------------
__global__ __launch_bounds__(TPB)
void ohlcv_partial_kernel(const float* __restrict__ target,
                          const int*   __restrict__ obs,      // observed_mask
                          const int*   __restrict__ sample_id,
                          const int*   __restrict__ variate_id,
                          float*       __restrict__ ws) {
  __shared__ int   sbin[TPB];
  __shared__ float sstat[TPB * 3];

  const int  tid = threadIdx.x;
  const long t   = (long)blockIdx.x * TPB + tid;     // global token index

  // --- per-token reduction over D=32 (b128 vector loads, fixed order) ---
  const float4* tp = (const float4*)(target + t * DIM);
  const int4*   mp = (const int4*)(obs + t * DIM);
  float cnt = 0.f, s1 = 0.f, s2 = 0.f;
#pragma unroll
  for (int q = 0; q < 8; ++q) {
    float4 v = tp[q];
    int4   m = mp[q];
    float f0 = (m.x != 0) ? 1.f : 0.f;
    float f1 = (m.y != 0) ? 1.f : 0.f;
    float f2 = (m.z != 0) ? 1.f : 0.f;
    float f3 = (m.w != 0) ? 1.f : 0.f;
    cnt += (f0 + f1) + (f2 + f3);
    s1 = fmaf(f0, v.x, s1); s1 = fmaf(f1, v.y, s1);
    s1 = fmaf(f2, v.z, s1); s1 = fmaf(f3, v.w, s1);
    s2 = fmaf(f0 * v.x, v.x, s2); s2 = fmaf(f1 * v.y, v.y, s2);
    s2 = fmaf(f2 * v.z, v.z, s2); s2 = fmaf(f3 * v.w, v.w, s2);
  }

  const int sid = sample_id[t];
  const int vid = variate_id[t];
  sbin[tid]          = (sid * 4 + group_of(vid)) & (NBIN - 1);
  sstat[tid * 3 + 0] = cnt;
  sstat[tid * 3 + 1] = s1;
  sstat[tid * 3 + 2] = s2;
  __syncthreads();

  // 96 lanes: one (bin, field) each; fixed j=0..255 order => deterministic.
  if (tid < NBIN * 3) {
    const int g = tid / 3;
    const int k = tid - 3 * g;
    float acc = 0.f;
    for (int j = 0; j < TPB; ++j) {
      acc += (sbin[j] == g) ? sstat[j * 3 + k] : 0.f;
    }
    ws[((long)blockIdx.x * NBIN + g) * BIN_STRIDE + k] = acc;
  }
}

// ---------------------------------------------------------------------------
extern "C" void kernel_launch(void* const* d_in, const int* in_sizes, int n_in,
                              void* d_out, int out_size, void* d_ws,
                              size_t ws_size, hipStream_t stream) {
  (void)in_sizes; (void)n_in; (void)out_size; (void)ws_size;
  const float* target     = (const float*)d_in[0];
  const int*   obs        = (const int*)d_in[1];   // bool -> int32 per harness
  const int*   sample_id  = (const int*)d_in[2];
  const int*   variate_id = (const int*)d_in[3];
  float*       ws         = (float*)d_ws;          // needs 32 KB
  float*       out        = (float*)d_out;

  ohlcv_partial_kernel<<<TOTAL_BLOCKS, TPB, 0, stream>>>(
      target, obs, sample_id, variate_id, ws);
  ohlcv_finalize_kernel<<<TOTAL_BLOCKS, TPB, 0, stream>>>(
      ws, sample_id, variate_id, out);
}